// BLR_69458211111223
// MI455X (gfx1250) — compile-verified
//
#include <hip/hip_runtime.h>
#include <math.h>

typedef __attribute__((ext_vector_type(2))) float v2f;
typedef __attribute__((ext_vector_type(4))) float v4f;
typedef __attribute__((ext_vector_type(8))) float v8f;

// D = A(16x4) * B(4x16) + C, fp32 WMMA.
__device__ __forceinline__ v8f wmma_k4(v2f a, v2f b, v8f c) {
  return __builtin_amdgcn_wmma_f32_16x16x4_f32(false, a, false, b, (short)0, c,
                                               false, false);
}

// Async global -> LDS copy (ASYNCcnt tracked), CDNA5 VGLOBAL op.
__device__ __forceinline__ void async_g2l_b128(unsigned lds_addr,
                                               const void* gaddr) {
  asm volatile("global_load_async_to_lds_b128 %0, %1, off" ::"v"(lds_addr),
               "v"(gaddr)
               : "memory");
}
__device__ __forceinline__ void wait_async0() {
  asm volatile("s_wait_asynccnt 0" ::: "memory");
}

// ---------------------------------------------------------------------------
// Sizes
// ---------------------------------------------------------------------------
#define BATCH 128
#define NS    128
#define NQ    512
#define DIN   256
#define DOUT  64

#define MU_ELEMS  (128u * 512u * 64u)                 // 4194304
#define SIG_F4    ((size_t)128 * 512 * 64 * 16)       // 67108864 float4
#define NLL_OFF   ((size_t)4194304 + (size_t)268435456)

// workspace float offsets
#define OFF_PRIOR 0u
#define OFF_MNAT  65536u
#define OFF_S     81920u
#define OFF_RHS   8470528u
#define OFF_M     10567680u
#define OFF_SPR   12664832u
#define OFF_QS    12730368u

// ---------------------------------------------------------------------------
// K0a: S_inv_prior = asym @ asym^T    (256x256, K=256)
// ---------------------------------------------------------------------------
__global__ __launch_bounds__(256) void k_prior_prec(
    const float* __restrict__ asym, float* __restrict__ Spr) {
  int lane = threadIdx.x & 31;
  int wave = (blockIdx.x << 3) + (threadIdx.x >> 5);  // 0..255
  int tm = (wave >> 4) << 4, tn = (wave & 15) << 4;
  int lm = lane & 15, ko = (lane >> 4) << 1, hi8 = (lane >> 4) << 3;
  v8f acc = {0.f, 0.f, 0.f, 0.f, 0.f, 0.f, 0.f, 0.f};
  for (int k = 0; k < 256; k += 4) {
    const float* pa = asym + (tm + lm) * 256 + (k + ko);
    const float* pb = asym + (tn + lm) * 256 + (k + ko);
    v2f a = {pa[0], pa[1]};
    v2f b = {pb[0], pb[1]};
    acc = wmma_k4(a, b, acc);
  }
#pragma unroll
  for (int r = 0; r < 8; ++r)
    Spr[(tm + hi8 + r) * 256 + tn + lm] = acc[r];
}

// ---------------------------------------------------------------------------
// K0b: m_prior_nat = S_inv_prior @ m_prior   (256x64, K=256)
// ---------------------------------------------------------------------------
__global__ __launch_bounds__(256) void k_prior_mean(
    const float* __restrict__ Spr, const float* __restrict__ mp,
    float* __restrict__ mnat) {
  int lane = threadIdx.x & 31;
  int wave = (blockIdx.x << 3) + (threadIdx.x >> 5);  // 0..63
  int tm = (wave >> 2) << 4, tn = (wave & 3) << 4;
  int lm = lane & 15, ko = (lane >> 4) << 1, hi8 = (lane >> 4) << 3;
  v8f acc = {0.f, 0.f, 0.f, 0.f, 0.f, 0.f, 0.f, 0.f};
  for (int k = 0; k < 256; k += 4) {
    v2f a = {Spr[(tm + lm) * 256 + k + ko], Spr[(tm + lm) * 256 + k + ko + 1]};
    v2f b = {mp[(k + ko) * 64 + tn + lm], mp[(k + ko + 1) * 64 + tn + lm]};
    acc = wmma_k4(a, b, acc);
  }
#pragma unroll
  for (int r = 0; r < 8; ++r)
    mnat[(tm + hi8 + r) * 64 + tn + lm] = acc[r];
}

// ---------------------------------------------------------------------------
// K1: per batch  S_inv = phi_s^T phi_s + prior ;  rhs = phi_s^T y_s + mnat
// grid (40, 128): blocks 0..31 -> S tiles, 32..39 -> rhs tiles (WMMA branch).
// ---------------------------------------------------------------------------
__global__ __launch_bounds__(256) void k_gram(
    const float* __restrict__ phis, const float* __restrict__ ys,
    const float* __restrict__ Spr, const float* __restrict__ mnat,
    float* __restrict__ Sinv, float* __restrict__ rhs) {
  int b = blockIdx.y;
  int lane = threadIdx.x & 31;
  int t = (blockIdx.x << 3) + (threadIdx.x >> 5);  // 0..319
  int lm = lane & 15, ko = (lane >> 4) << 1, hi8 = (lane >> 4) << 3;
  const float* phib = phis + (size_t)b * NS * DIN;
  if (blockIdx.x < 32) {
    int tm = (t >> 4) << 4, tn = (t & 15) << 4;
    v8f acc;
#pragma unroll
    for (int r = 0; r < 8; ++r) acc[r] = Spr[(tm + hi8 + r) * 256 + tn + lm];
    for (int k = 0; k < NS; k += 4) {
      v2f a = {phib[(k + ko) * 256 + tm + lm],
               phib[(k + ko + 1) * 256 + tm + lm]};
      v2f bb = {phib[(k + ko) * 256 + tn + lm],
                phib[(k + ko + 1) * 256 + tn + lm]};
      acc = wmma_k4(a, bb, acc);
    }
    float* Sb = Sinv + (size_t)b * 65536;
#pragma unroll
    for (int r = 0; r < 8; ++r)
      Sb[(tm + hi8 + r) * 256 + tn + lm] = acc[r];
  } else {
    int tt = t - 256;
    int tm = (tt >> 2) << 4, tn = (tt & 3) << 4;
    const float* yb = ys + (size_t)b * NS * DOUT;
    v8f acc;
#pragma unroll
    for (int r = 0; r < 8; ++r) acc[r] = mnat[(tm + hi8 + r) * 64 + tn + lm];
    for (int k = 0; k < NS; k += 4) {
      v2f a = {phib[(k + ko) * 256 + tm + lm],
               phib[(k + ko + 1) * 256 + tm + lm]};
      v2f bb = {yb[(k + ko) * 64 + tn + lm], yb[(k + ko + 1) * 64 + tn + lm]};
      acc = wmma_k4(a, bb, acc);
    }
    float* rb = rhs + (size_t)b * 16384;
#pragma unroll
    for (int r = 0; r < 8; ++r)
      rb[(tm + hi8 + r) * 64 + tn + lm] = acc[r];
  }
}

// ---------------------------------------------------------------------------
// K2: blocked (nb=16) in-place Gauss-Jordan inverse in LDS with WMMA
// rank-16 trailing updates, then fused m = S @ rhs.
// 1 block/batch, 1024 threads (32 waves), ~285 KB LDS.
// Per block step kb:
//   P = A[kb,kb];  Pinv = inv(P)            (single-wave lockstep GJ)
//   Fneg = -A[:, kb-cols]                   (snapshot)
//   row panel:  A[kb, j] = Pinv @ A[kb, j]  (j != kb), A[kb,kb] = Pinv
//   trailing :  A[i, j] += Fneg_i @ A[kb, j] (i != kb; C=0 when j == kb)
// ---------------------------------------------------------------------------
#define LST 257
#define CST 20
#define PST 17
__global__ __launch_bounds__(1024) void k_invert(
    float* __restrict__ Sinv, const float* __restrict__ rhs,
    float* __restrict__ mout) {
  __shared__ float As[256 * LST];
  __shared__ float Fneg[256 * CST];
  __shared__ float Pb[16 * PST];
  int b = blockIdx.x;
  int tid = threadIdx.x;
  int lane = tid & 31;
  int wave = tid >> 5;  // 0..31
  int lm = lane & 15, ko = (lane >> 4) << 1, hi8 = (lane >> 4) << 3;
  float* Sb = Sinv + (size_t)b * 65536;
  for (int e = tid; e < 65536; e += 1024)
    As[(e >> 8) * LST + (e & 255)] = Sb[e];
  __syncthreads();

  for (int kb = 0; kb < 16; ++kb) {
    int k0 = kb << 4;
    // phase 1: snapshot pivot block and negated column panel
    if (tid < 256)
      Pb[(tid >> 4) * PST + (tid & 15)] =
          As[(k0 + (tid >> 4)) * LST + k0 + (tid & 15)];
    for (int e = tid; e < 4096; e += 1024) {
      int r = e >> 4, c = e & 15;
      Fneg[r * CST + c] = -As[r * LST + k0 + c];
    }
    __syncthreads();
    // phase 2: invert Pb in place; single wave => lockstep, no barriers
    if (wave == 0) {
      int pi = lane >> 1;
      int j0 = (lane & 1) << 3;
      for (int k = 0; k < 16; ++k) {
        float pv = 1.0f / Pb[k * PST + k];
        float f = Pb[pi * PST + k];
        float rk[8], pij[8];
#pragma unroll
        for (int jj = 0; jj < 8; ++jj) {
          rk[jj] = Pb[k * PST + j0 + jj];
          pij[jj] = Pb[pi * PST + j0 + jj];
        }
#pragma unroll
        for (int jj = 0; jj < 8; ++jj) {
          int j = j0 + jj;
          float rs = rk[jj] * pv;
          float nv;
          if (pi == k)
            nv = (j == k) ? pv : rs;
          else
            nv = (j == k) ? (-f * pv) : (pij[jj] - f * rs);
          Pb[pi * PST + j] = nv;
        }
      }
    }
    __syncthreads();
    // phase 3: row panel, waves 0..15 only (in-place Rold -> Rnew)
    if (wave < 16) {
      int tn = wave << 4;
      if (wave == kb) {  // A[kb,kb] = Pinv
        for (int e = lane; e < 256; e += 32)
          As[(k0 + (e >> 4)) * LST + k0 + (e & 15)] =
              Pb[(e >> 4) * PST + (e & 15)];
      } else {
        v8f acc = {0.f, 0.f, 0.f, 0.f, 0.f, 0.f, 0.f, 0.f};
#pragma unroll
        for (int ks = 0; ks < 16; ks += 4) {
          v2f a = {Pb[lm * PST + ks + ko], Pb[lm * PST + ks + ko + 1]};
          v2f bb = {As[(k0 + ks + ko) * LST + tn + lm],
                    As[(k0 + ks + ko + 1) * LST + tn + lm]};
          acc = wmma_k4(a, bb, acc);
        }
#pragma unroll
        for (int r = 0; r < 8; ++r)
          As[(k0 + hi8 + r) * LST + tn + lm] = acc[r];
      }
    }
    __syncthreads();
    // phase 4: trailing rank-16 update, 240 tiles over 32 waves
    for (int it = 0; it < 8; ++it) {
      int t = wave + (it << 5);
      int tmb = t >> 4, tnb = t & 15;
      if (tmb == kb) continue;  // pivot rows are final
      int tm = tmb << 4, tn = tnb << 4;
      bool zc = (tnb == kb);  // column panel: pure -F @ Pinv
      v8f acc;
#pragma unroll
      for (int r = 0; r < 8; ++r)
        acc[r] = zc ? 0.0f : As[(tm + hi8 + r) * LST + tn + lm];
#pragma unroll
      for (int ks = 0; ks < 16; ks += 4) {
        v2f a = {Fneg[(tm + lm) * CST + ks + ko],
                 Fneg[(tm + lm) * CST + ks + ko + 1]};
        v2f bb = {As[(k0 + ks + ko) * LST + tn + lm],
                  As[(k0 + ks + ko + 1) * LST + tn + lm]};
        acc = wmma_k4(a, bb, acc);
      }
#pragma unroll
      for (int r = 0; r < 8; ++r)
        As[(tm + hi8 + r) * LST + tn + lm] = acc[r];
    }
    __syncthreads();
  }

  // write S = inv(S_inv) back
  for (int e = tid; e < 65536; e += 1024)
    Sb[e] = As[(e >> 8) * LST + (e & 255)];
  // m = S @ rhs : 64 tiles over 32 waves, A-frags straight from LDS
  const float* rb = rhs + (size_t)b * 16384;
  float* mb = mout + (size_t)b * 16384;
#pragma unroll
  for (int half = 0; half < 2; ++half) {
    int t = wave * 2 + half;  // 0..63
    int tm = (t >> 2) << 4, tn = (t & 3) << 4;
    v8f acc = {0.f, 0.f, 0.f, 0.f, 0.f, 0.f, 0.f, 0.f};
    for (int k = 0; k < 256; k += 4) {
      v2f a = {As[(tm + lm) * LST + k + ko], As[(tm + lm) * LST + k + ko + 1]};
      v2f bb = {rb[(k + ko) * 64 + tn + lm], rb[(k + ko + 1) * 64 + tn + lm]};
      acc = wmma_k4(a, bb, acc);
    }
#pragma unroll
    for (int r = 0; r < 8; ++r)
      mb[(tm + hi8 + r) * 64 + tn + lm] = acc[r];
  }
}

// ---------------------------------------------------------------------------
// K4: per 16-query tile: stage phi_q tile in LDS via ASYNC global->LDS B128,
// WMMA t = phi_q @ S consumed on-the-fly into spread, WMMA mu = phi_q @ m
// with fused residual accumulation. grid (4,128), 256 threads (8 waves).
// ---------------------------------------------------------------------------
#define QST 260
__global__ __launch_bounds__(256) void k_query(
    const float* __restrict__ phiq, const float* __restrict__ yq,
    const float* __restrict__ S, const float* __restrict__ m,
    float* __restrict__ mu, float* __restrict__ spread_ws,
    float* __restrict__ qsum_ws) {
  __shared__ __align__(16) float slab_all[8][16 * QST];
  int b = blockIdx.y;
  int lane = threadIdx.x & 31;
  int wave = threadIdx.x >> 5;
  int qt = (blockIdx.x << 3) + wave;  // 0..31
  int q0 = qt << 4;
  float* slab = slab_all[wave];
  unsigned slab_lds = (unsigned)(size_t)slab;  // low 32 bits = LDS offset
  const float* pq = phiq + ((size_t)b * NQ + q0) * DIN;
  // async-stage 16x256 tile; row pitch 260 floats (1040 B, 16-aligned)
  for (int it = 0; it < 32; ++it) {
    int q4 = lane + (it << 5);  // float4 index 0..1023
    int r = q4 >> 6;
    int c4 = q4 & 63;
    unsigned dst = slab_lds + (unsigned)((r * QST + (c4 << 2)) << 2);
    async_g2l_b128(dst, pq + ((size_t)q4 << 2));
  }
  wait_async0();
  __syncthreads();
  int lm = lane & 15, ko = (lane >> 4) << 1, hi8 = (lane >> 4) << 3;
  const float* Sb = S + (size_t)b * 65536;
  // ---- spread = 1 + diag(phi_q S phi_q^T) ----
  float sacc[8] = {0.f, 0.f, 0.f, 0.f, 0.f, 0.f, 0.f, 0.f};
  for (int ntile = 0; ntile < 16; ++ntile) {
    int nt = ntile << 4;
    v8f acc = {0.f, 0.f, 0.f, 0.f, 0.f, 0.f, 0.f, 0.f};
    for (int k = 0; k < 256; k += 4) {
      v2f a = {slab[lm * QST + k + ko], slab[lm * QST + k + ko + 1]};
      v2f bb = {Sb[(k + ko) * 256 + nt + lm], Sb[(k + ko + 1) * 256 + nt + lm]};
      acc = wmma_k4(a, bb, acc);
    }
#pragma unroll
    for (int r = 0; r < 8; ++r)
      sacc[r] += acc[r] * slab[(hi8 + r) * QST + nt + lm];
  }
#pragma unroll
  for (int r = 0; r < 8; ++r) {
    float v = sacc[r];
    v += __shfl_xor(v, 1, 32);
    v += __shfl_xor(v, 2, 32);
    v += __shfl_xor(v, 4, 32);
    v += __shfl_xor(v, 8, 32);
    sacc[r] = v;
  }
  if (lm == 0) {
#pragma unroll
    for (int r = 0; r < 8; ++r)
      spread_ws[(size_t)b * NQ + q0 + hi8 + r] = 1.0f + sacc[r];
  }
  // ---- mu = phi_q @ m, fused residual^2 ----
  const float* mb = m + (size_t)b * 16384;
  const float* yb = yq + ((size_t)b * NQ + q0) * DOUT;
  float* mub = mu + ((size_t)b * NQ + q0) * DOUT;
  float racc[8] = {0.f, 0.f, 0.f, 0.f, 0.f, 0.f, 0.f, 0.f};
  for (int ntile = 0; ntile < 4; ++ntile) {
    int nt = ntile << 4;
    v8f acc = {0.f, 0.f, 0.f, 0.f, 0.f, 0.f, 0.f, 0.f};
    for (int k = 0; k < 256; k += 4) {
      v2f a = {slab[lm * QST + k + ko], slab[lm * QST + k + ko + 1]};
      v2f bb = {mb[(k + ko) * 64 + nt + lm], mb[(k + ko + 1) * 64 + nt + lm]};
      acc = wmma_k4(a, bb, acc);
    }
#pragma unroll
    for (int r = 0; r < 8; ++r) {
      int row = hi8 + r;
      float mv = acc[r];
      mub[row * 64 + nt + lm] = mv;
      float res = yb[row * 64 + nt + lm] - mv;
      racc[r] += res * res;
    }
  }
#pragma unroll
  for (int r = 0; r < 8; ++r) {
    float v = racc[r];
    v += __shfl_xor(v, 1, 32);
    v += __shfl_xor(v, 2, 32);
    v += __shfl_xor(v, 4, 32);
    v += __shfl_xor(v, 8, 32);
    racc[r] = v;
  }
  if (lm == 0) {
#pragma unroll
    for (int r = 0; r < 8; ++r)
      qsum_ws[(size_t)b * NQ + q0 + hi8 + r] = racc[r];
  }
}

// ---------------------------------------------------------------------------
// K5: sig diag fill, 1 GB streamed with NON-TEMPORAL float4 stores so the
// write stream does not evict S / phi working set from L2.
// ---------------------------------------------------------------------------
__global__ __launch_bounds__(256) void k_sig(
    float* __restrict__ sig, const float* __restrict__ spread,
    const float* __restrict__ sig_eps) {
  size_t idx = (size_t)blockIdx.x * 256 + threadIdx.x;  // one float4 each
  float se = sig_eps[0];
  size_t t = idx;
  int j4 = (int)(t & 15);
  t >>= 4;
  int i = (int)(t & 63);
  t >>= 6;
  size_t bq = t;  // 0..65535
  float d = spread[bq] * se;
  int j0 = j4 << 2;
  v4f v = {(i == j0) ? d : 0.f, (i == j0 + 1) ? d : 0.f,
           (i == j0 + 2) ? d : 0.f, (i == j0 + 3) ? d : 0.f};
  __builtin_nontemporal_store(v, (v4f*)sig + idx);
}

// ---------------------------------------------------------------------------
// K6: nll = mean(64*(log(spread)+log(se))) + mean(qsum/(spread*se))
// ---------------------------------------------------------------------------
__global__ __launch_bounds__(1024) void k_nll(
    const float* __restrict__ spread, const float* __restrict__ qsum,
    const float* __restrict__ sig_eps, float* __restrict__ out) {
  __shared__ float red[1024];
  int tid = threadIdx.x;
  float se = sig_eps[0];
  float lse = logf(se);
  float acc = 0.f;
  for (int e = tid; e < 65536; e += 1024) {
    float sp = spread[e];
    acc += 64.0f * (logf(sp) + lse) + qsum[e] / (sp * se);
  }
  red[tid] = acc;
  __syncthreads();
  for (int s = 512; s > 0; s >>= 1) {
    if (tid < s) red[tid] += red[tid + s];
    __syncthreads();
  }
  if (tid == 0) out[0] = red[0] * (1.0f / 65536.0f);
}

// ---------------------------------------------------------------------------
extern "C" void kernel_launch(void* const* d_in, const int* in_sizes, int n_in,
                              void* d_out, int out_size, void* d_ws,
                              size_t ws_size, hipStream_t stream) {
  (void)in_sizes;
  (void)n_in;
  (void)out_size;
  (void)ws_size;
  const float* phi_s = (const float*)d_in[0];
  const float* y_s = (const float*)d_in[1];
  const float* phi_q = (const float*)d_in[2];
  const float* y_q = (const float*)d_in[3];
  const float* m_pr = (const float*)d_in[4];
  const float* asym = (const float*)d_in[5];
  const float* s_eps = (const float*)d_in[6];
  float* out = (float*)d_out;
  float* ws = (float*)d_ws;
  float* wsPrior = ws + OFF_PRIOR;
  float* wsMnat = ws + OFF_MNAT;
  float* wsS = ws + OFF_S;
  float* wsRhs = ws + OFF_RHS;
  float* wsM = ws + OFF_M;
  float* wsSpr = ws + OFF_SPR;
  float* wsQs = ws + OFF_QS;
  float* mu = out;
  float* sig = out + MU_ELEMS;
  float* nll = out + NLL_OFF;

  k_prior_prec<<<32, 256, 0, stream>>>(asym, wsPrior);
  k_prior_mean<<<8, 256, 0, stream>>>(wsPrior, m_pr, wsMnat);
  k_gram<<<dim3(40, 128), 256, 0, stream>>>(phi_s, y_s, wsPrior, wsMnat, wsS,
                                            wsRhs);
  k_invert<<<128, 1024, 0, stream>>>(wsS, wsRhs, wsM);
  k_query<<<dim3(4, 128), 256, 0, stream>>>(phi_q, y_q, wsS, wsM, mu, wsSpr,
                                            wsQs);
  k_sig<<<(unsigned)(SIG_F4 / 256), 256, 0, stream>>>(sig, wsSpr, s_eps);
  k_nll<<<1, 1024, 0, stream>>>(wsSpr, wsQs, s_eps, nll);
}